// FNODEQ_56384330662285
// MI455X (gfx1250) — compile-verified
//
#include <hip/hip_runtime.h>
#include <math.h>

// ---------------- constants ----------------
constexpr int   Bn  = 4;
constexpr int   Cc  = 32;
constexpr int   Hh  = 192;
constexpr int   Wp  = 192;
constexpr int   Pp  = Hh * Wp;            // 36864 pixels per (b,c) plane
constexpr long  dB  = (long)Cc * Pp;      // 1,179,648 elems per batch
constexpr long  dT  = (long)Bn * dB;      // 4,718,592 elems total per z tensor
constexpr int   NM  = 12;                 // kept modes per dim (M1=M2=12)
constexpr int   NKX = 24;                 // 12 top + 12 bottom kx rows
constexpr int   GRAM_CH = 144;            // gram reduction chunks (dB/144 = 8192)
constexpr float EPSV = 1e-5f;
constexpr float PI2_192 = 6.28318530717958647692f / 192.0f;

typedef float v2f __attribute__((ext_vector_type(2)));
typedef float v8f __attribute__((ext_vector_type(8)));

#define DEV __device__ __forceinline__

DEV float gelu_f(float x) { return 0.5f * x * (1.0f + erff(x * 0.70710678118654752440f)); }
DEV float sigm_f(float x) { return 1.0f / (1.0f + expf(-x)); }

DEV float blk_sum256(float v) {
  __shared__ float sm[256];
  int t = threadIdx.x;
  __syncthreads();                 // protect any previous use of sm
  sm[t] = v; __syncthreads();
  #pragma unroll
  for (int s = 128; s > 0; s >>= 1) {
    if (t < s) sm[t] += sm[t + s];
    __syncthreads();
  }
  return sm[0];
}

// ---------------- trivial utility kernels ----------------
__global__ void k_fill0(float* p, long n) {
  long t = (long)blockIdx.x * blockDim.x + threadIdx.x;
  if (t < n) p[t] = 0.f;
}
__global__ void k_copy(const float* __restrict__ s, float* __restrict__ d, long n) {
  long t = (long)blockIdx.x * blockDim.x + threadIdx.x;
  if (t < n) d[t] = s[t];
}

// DFT twiddle tables: cwt/swt[ky*192+w] = cos/sin(2pi*w*ky/192)   (ky<12)
//                     cht/sht[kxi*192+h] = cos/sin(2pi*h*kx/192)  (kx = kxi or 168+kxi)
__global__ void k_tables(float* cwt, float* swt, float* cht, float* sht) {
  int t = blockIdx.x * blockDim.x + threadIdx.x;
  if (t >= NKX * 192) return;
  int k = t / 192, idx = t % 192;
  int kx = (k < NM) ? k : (168 + k);      // 12..23 -> 180..191
  int ph = (idx * kx) % 192;
  float a = (float)ph * PI2_192;
  cht[t] = cosf(a); sht[t] = sinf(a);
  if (k < NM) {
    int ph2 = (idx * k) % 192;
    float a2 = (float)ph2 * PI2_192;
    cwt[t] = cosf(a2); swt[t] = sinf(a2);
  }
}

// ---------------- lift + injection (once per call) ----------------
__global__ void k_fc0(const float* __restrict__ x, const float* __restrict__ w,
                      const float* __restrict__ b0, float* __restrict__ xin) {
  long t = (long)blockIdx.x * blockDim.x + threadIdx.x;
  if (t >= dT) return;
  long p = t % Pp; long c = (t / Pp) % Cc; long b = t / ((long)Pp * Cc);
  xin[t] = x[b * Pp + p] * w[c] + b0[c];
}

__global__ void k_conv3(const float* __restrict__ xin, const float* __restrict__ cw,
                        float* __restrict__ u) {
  long t = (long)blockIdx.x * blockDim.x + threadIdx.x;
  if (t >= (long)Bn * 16 * Pp) return;
  int w2 = t % Wp; int h = (t / Wp) % Hh; int o = (t / Pp) % 16; int b = t / ((long)Pp * 16);
  float a = 0.f;
  for (int i = 0; i < Cc; ++i) {
    const float* base = xin + ((long)b * Cc + i) * Pp;
    const float* kw = cw + ((long)o * Cc + i) * 9;
    #pragma unroll
    for (int kh = 0; kh < 3; ++kh) {
      int hh = h + kh - 1; if (hh < 0 || hh >= Hh) continue;
      #pragma unroll
      for (int kx = 0; kx < 3; ++kx) {
        int ww = w2 + kx - 1; if (ww < 0 || ww >= Wp) continue;
        a += base[hh * Wp + ww] * kw[kh * 3 + kx];
      }
    }
  }
  u[t] = a;
}

// group-norm stats: one block per (b, group)
__global__ void k_gn_stats(const float* __restrict__ src, int chPerGroup, int nGroups,
                           float* gm, float* gr) {
  int bg = blockIdx.x; int g = bg % nGroups; int b = bg / nGroups;
  long base = ((long)b * nGroups * chPerGroup + (long)g * chPerGroup) * Pp;
  long cnt  = (long)chPerGroup * Pp;
  float s = 0.f, s2 = 0.f;
  for (long e = threadIdx.x; e < cnt; e += 256) {
    float v = src[base + e]; s += v; s2 += v * v;
  }
  float S = blk_sum256(s);
  float S2 = blk_sum256(s2);
  if (threadIdx.x == 0) {
    float mu = S / (float)cnt;
    float var = S2 / (float)cnt - mu * mu;
    gm[bg] = mu; gr[bg] = rsqrtf(var + EPSV);
  }
}

// xi = sigmoid( q_w @ groupnorm(conv_out) + q_b )
__global__ void k_injq(const float* __restrict__ u, const float* __restrict__ gm,
                       const float* __restrict__ gr, const float* __restrict__ iw,
                       const float* __restrict__ ib, const float* __restrict__ qw,
                       const float* __restrict__ qb, float* __restrict__ xi) {
  long t = (long)blockIdx.x * blockDim.x + threadIdx.x;
  if (t >= dT) return;
  long p = t % Pp; int d = (t / Pp) % Cc; int b = t / ((long)Pp * Cc);
  float a = qb[d];
  #pragma unroll
  for (int r = 0; r < 16; ++r) {
    int g = r >> 3;
    float m = gm[b * 2 + g], rs = gr[b * 2 + g];
    float uv = u[((long)b * 16 + r) * Pp + p];
    a += ((uv - m) * rs * iw[r] + ib[r]) * qw[d * 16 + r];
  }
  xi[t] = sigm_f(a);
}

// ---------------- truncated spectral conv ----------------
// Zy[b,c,h,ky]{re,im} = sum_w z * e^{-2pi i w ky/192}, ky<12
__global__ void k_dftw(const float* __restrict__ z, const float* __restrict__ cwt,
                       const float* __restrict__ swt, float* __restrict__ Zy) {
  int t = blockIdx.x * blockDim.x + threadIdx.x;
  if (t >= Bn * Cc * Hh * NM) return;
  int ky = t % NM; int h = (t / NM) % Hh; int ch = t / (NM * Hh);
  const float* row = z + ((long)ch * Hh + h) * Wp;
  const float* cw = cwt + ky * Wp;
  const float* sw = swt + ky * Wp;
  float re = 0.f, im = 0.f;
  for (int w = 0; w < Wp; ++w) { float v = row[w]; re += v * cw[w]; im -= v * sw[w]; }
  long o = ((long)ch * Hh + h) * NM + ky;
  Zy[o * 2] = re; Zy[o * 2 + 1] = im;
}

// Zf[b,c,kxi,ky] = sum_h Zy * e^{-2pi i h kx/192}
__global__ void k_dfth(const float* __restrict__ Zy, const float* __restrict__ cht,
                       const float* __restrict__ sht, float* __restrict__ Zf) {
  int t = blockIdx.x * blockDim.x + threadIdx.x;
  if (t >= Bn * Cc * NKX * NM) return;
  int ky = t % NM; int kxi = (t / NM) % NKX; int ch = t / (NM * NKX);
  float re = 0.f, im = 0.f;
  for (int h = 0; h < Hh; ++h) {
    long zi = (((long)ch * Hh + h) * NM + ky) * 2;
    float zr = Zy[zi], zim = Zy[zi + 1];
    float c = cht[kxi * Wp + h], s = sht[kxi * Wp + h];
    re += zr * c + zim * s;      // (zr+i zi)(c - i s)
    im += zim * c - zr * s;
  }
  long o = (((long)ch * NKX + kxi) * NM + ky) * 2;
  Zf[o] = re; Zf[o + 1] = im;
}

// O[b,o,kxi,ky] = sum_i Zf[b,i,kxi,ky] * W[i,o,x,ky]   (complex)
__global__ void k_mix(const float* __restrict__ Zf,
                      const float* __restrict__ w1r, const float* __restrict__ w1i,
                      const float* __restrict__ w2r, const float* __restrict__ w2i,
                      float* __restrict__ Om) {
  int t = blockIdx.x * blockDim.x + threadIdx.x;
  if (t >= Bn * Cc * NKX * NM) return;
  int ky = t % NM; int kxi = (t / NM) % NKX; int o = (t / (NM * NKX)) % Cc;
  int b = t / (NM * NKX * Cc);
  int x = (kxi < NM) ? kxi : (kxi - NM);
  const float* wr = (kxi < NM) ? w1r : w2r;
  const float* wi = (kxi < NM) ? w1i : w2i;
  float re = 0.f, im = 0.f;
  for (int i = 0; i < Cc; ++i) {
    long zi = ((((long)b * Cc + i) * NKX + kxi) * NM + ky) * 2;
    float zr = Zf[zi], zim = Zf[zi + 1];
    long widx = (((long)i * Cc + o) * NM + x) * NM + ky;
    float wre = wr[widx], wim = wi[widx];
    re += zr * wre - zim * wim;
    im += zr * wim + zim * wre;
  }
  long oo = ((((long)b * Cc + o) * NKX + kxi) * NM + ky) * 2;
  Om[oo] = re; Om[oo + 1] = im;
}

// T[b,o,kxi,w] = sum_ky f_ky * O * e^{+2pi i ky w/192}; f_0=1 else 2
__global__ void k_synw(const float* __restrict__ Om, const float* __restrict__ cwt,
                       const float* __restrict__ swt, float* __restrict__ T) {
  int t = blockIdx.x * blockDim.x + threadIdx.x;
  if (t >= Bn * Cc * NKX * Wp) return;
  int w = t % Wp; int kxi = (t / Wp) % NKX; int bo = t / (Wp * NKX);
  float re = 0.f, im = 0.f;
  #pragma unroll
  for (int ky = 0; ky < NM; ++ky) {
    long oi = (((long)bo * NKX + kxi) * NM + ky) * 2;
    float Or = Om[oi], Oi = Om[oi + 1];
    float c = cwt[ky * Wp + w], s = swt[ky * Wp + w];
    float f = (ky == 0) ? 1.f : 2.f;
    re += f * (Or * c - Oi * s);
    im += f * (Or * s + Oi * c);
  }
  long to = (((long)bo * NKX + kxi) * Wp + w) * 2;
  T[to] = re; T[to + 1] = im;
}

// t1 = xi + gelu( (1/(H*W)) * sum_kx Re{T e^{+2pi i kx h/192}} + y2 )
__global__ void k_synh(const float* __restrict__ T, const float* __restrict__ cht,
                       const float* __restrict__ sht, const float* __restrict__ y2,
                       const float* __restrict__ xi, float* __restrict__ t1) {
  long t = (long)blockIdx.x * blockDim.x + threadIdx.x;
  if (t >= dT) return;
  int w = t % Wp; int h = (t / Wp) % Hh; long bo = t / Pp;
  float acc = 0.f;
  #pragma unroll
  for (int kxi = 0; kxi < NKX; ++kxi) {
    long ti = (((long)bo * NKX + kxi) * Wp + w) * 2;
    acc += T[ti] * cht[kxi * Wp + h] - T[ti + 1] * sht[kxi * Wp + h];
  }
  float y1 = acc * (1.0f / (float)(Hh * Wp));
  t1[t] = xi[t] + gelu_f(y1 + y2[t]);
}

// ---------------- WMMA f32 GEMMs: 32ch -> 32ch over pixels ----------------
// one wave: 32(M) x 16(N) output tile, K=32 in 8 steps of V_WMMA_F32_16X16X4_F32
__global__ void k_lin32_wmma(const float* __restrict__ Z, const float* __restrict__ Wm,
                             const float* __restrict__ bias, float* __restrict__ Y) {
  int wave = (blockIdx.x * blockDim.x + threadIdx.x) >> 5;
  int lane = threadIdx.x & 31;
  int half = lane >> 4;
  int l = lane & 15;
  int b = wave / (Pp / 16);
  int nBase = (wave % (Pp / 16)) * 16;
  const float* Zb = Z + (long)b * dB;
  float* Yb = Y + (long)b * dB;
  v8f c0 = {0.f, 0.f, 0.f, 0.f, 0.f, 0.f, 0.f, 0.f};
  v8f c1 = {0.f, 0.f, 0.f, 0.f, 0.f, 0.f, 0.f, 0.f};
  #pragma unroll
  for (int kk = 0; kk < 8; ++kk) {
    int k0 = kk * 4 + 2 * half;
    v2f bv; bv[0] = Zb[(long)k0 * Pp + nBase + l];
            bv[1] = Zb[(long)(k0 + 1) * Pp + nBase + l];
    v2f a0; a0[0] = Wm[l * Cc + k0];          a0[1] = Wm[l * Cc + k0 + 1];
    v2f a1; a1[0] = Wm[(16 + l) * Cc + k0];   a1[1] = Wm[(16 + l) * Cc + k0 + 1];
    c0 = __builtin_amdgcn_wmma_f32_16x16x4_f32(false, a0, false, bv, (short)0, c0, false, false);
    c1 = __builtin_amdgcn_wmma_f32_16x16x4_f32(false, a1, false, bv, (short)0, c1, false, false);
  }
  #pragma unroll
  for (int r = 0; r < 8; ++r) {
    int m0 = r + 8 * half;
    Yb[(long)m0 * Pp + nBase + l]        = c0[r] + bias[m0];
    Yb[(long)(16 + m0) * Pp + nBase + l] = c1[r] + bias[16 + m0];
  }
}

// head: BN folded into Fw/Fb; h1 = gelu(Fw@z + Fb); out = fc2_w . h1 + fc2_b
__global__ void k_head_wmma(const float* __restrict__ Z, const float* __restrict__ Fw,
                            const float* __restrict__ Fb, const float* __restrict__ W2,
                            const float* __restrict__ B2, float* __restrict__ Out) {
  int wave = (blockIdx.x * blockDim.x + threadIdx.x) >> 5;
  int lane = threadIdx.x & 31;
  int half = lane >> 4;
  int l = lane & 15;
  int b = wave / (Pp / 16);
  int nBase = (wave % (Pp / 16)) * 16;
  const float* Zb = Z + (long)b * dB;
  v8f c0 = {0.f, 0.f, 0.f, 0.f, 0.f, 0.f, 0.f, 0.f};
  v8f c1 = {0.f, 0.f, 0.f, 0.f, 0.f, 0.f, 0.f, 0.f};
  #pragma unroll
  for (int kk = 0; kk < 8; ++kk) {
    int k0 = kk * 4 + 2 * half;
    v2f bv; bv[0] = Zb[(long)k0 * Pp + nBase + l];
            bv[1] = Zb[(long)(k0 + 1) * Pp + nBase + l];
    v2f a0; a0[0] = Fw[l * Cc + k0];          a0[1] = Fw[l * Cc + k0 + 1];
    v2f a1; a1[0] = Fw[(16 + l) * Cc + k0];   a1[1] = Fw[(16 + l) * Cc + k0 + 1];
    c0 = __builtin_amdgcn_wmma_f32_16x16x4_f32(false, a0, false, bv, (short)0, c0, false, false);
    c1 = __builtin_amdgcn_wmma_f32_16x16x4_f32(false, a1, false, bv, (short)0, c1, false, false);
  }
  float part = 0.f;
  #pragma unroll
  for (int r = 0; r < 8; ++r) {
    int m0 = r + 8 * half;
    part += gelu_f(c0[r] + Fb[m0]) * W2[m0];
    part += gelu_f(c1[r] + Fb[16 + m0]) * W2[16 + m0];
  }
  part += __shfl_xor(part, 16, 32);            // sum the two M-halves per column
  if (half == 0) Out[(long)b * Pp + nBase + l] = part + B2[0];
}

// ---------------- residual + group norms ----------------
__global__ void k_resid(const float* __restrict__ zin, const float* __restrict__ t1,
                        const float* __restrict__ gm, const float* __restrict__ gr,
                        const float* __restrict__ nw, const float* __restrict__ nb,
                        float* __restrict__ t2) {
  long t = (long)blockIdx.x * blockDim.x + threadIdx.x;
  if (t >= dT) return;
  int c = (t / Pp) % Cc; int b = t / ((long)Pp * Cc); int g = c >> 4;
  float v = zin[t] + ((t1[t] - gm[b * 2 + g]) * gr[b * 2 + g] * nw[c] + nb[c]);
  t2[t] = v > 0.f ? v : 0.f;
}

__global__ void k_gnapply(const float* __restrict__ t2, const float* __restrict__ gm,
                          const float* __restrict__ gr, const float* __restrict__ nw,
                          const float* __restrict__ nb, float* __restrict__ out) {
  long t = (long)blockIdx.x * blockDim.x + threadIdx.x;
  if (t >= dT) return;
  int c = (t / Pp) % Cc; int b = t / ((long)Pp * Cc); int g = c >> 4;
  out[t] = (t2[t] - gm[b * 2 + g]) * gr[b * 2 + g] * nw[c] + nb[c];
}

// ---------------- Anderson acceleration ----------------
__global__ void k_gram(const float* __restrict__ Fm, const float* __restrict__ X, int n,
                       float* __restrict__ partial) {
  int chunk = blockIdx.x; int b = blockIdx.y;
  const long len = dB / GRAM_CH;               // 8192
  long e0 = (long)chunk * len;
  float g[5];
  float acc[15];
  int np = n * (n + 1) / 2;
  for (int p = 0; p < 15; ++p) acc[p] = 0.f;
  for (long e = e0 + threadIdx.x; e < e0 + len; e += 256) {
    long idx = (long)b * dB + e;
    for (int r = 0; r < n; ++r) g[r] = Fm[(long)r * dT + idx] - X[(long)r * dT + idx];
    int p = 0;
    for (int i = 0; i < n; ++i)
      for (int j = i; j < n; ++j) acc[p++] += g[i] * g[j];
  }
  for (int p = 0; p < np; ++p) {
    float red = blk_sum256(acc[p]);
    if (threadIdx.x == 0) partial[((long)chunk * Bn + b) * 16 + p] = red;
  }
}

__global__ void k_solve(const float* __restrict__ partial, int n, float* __restrict__ alpha) {
  int b = threadIdx.x;
  if (b >= Bn) return;
  int np = n * (n + 1) / 2;
  float S[15];
  for (int p = 0; p < np; ++p) {
    float s = 0.f;
    for (int c = 0; c < GRAM_CH; ++c) s += partial[((long)c * Bn + b) * 16 + p];
    S[p] = s;
  }
  int N = n + 1;
  float Hm[36], y[6];
  for (int i = 0; i < N * N; ++i) Hm[i] = 0.f;
  for (int j = 1; j < N; ++j) { Hm[j] = 1.f; Hm[j * N] = 1.f; }
  int p = 0;
  for (int i = 0; i < n; ++i)
    for (int j = i; j < n; ++j) {
      float v = S[p++];
      Hm[(i + 1) * N + (j + 1)] = v; Hm[(j + 1) * N + (i + 1)] = v;
    }
  for (int i = 0; i < n; ++i) Hm[(i + 1) * N + (i + 1)] += 1e-4f;  // + lam*I
  for (int i = 0; i < N; ++i) y[i] = 0.f;
  y[0] = 1.f;
  // Gauss-Jordan with partial pivot (Hm[0][0]==0 forces pivoting)
  for (int col = 0; col < N; ++col) {
    int piv = col; float best = fabsf(Hm[col * N + col]);
    for (int r = col + 1; r < N; ++r) {
      float a = fabsf(Hm[r * N + col]);
      if (a > best) { best = a; piv = r; }
    }
    if (piv != col) {
      for (int c = 0; c < N; ++c) { float tm = Hm[col * N + c]; Hm[col * N + c] = Hm[piv * N + c]; Hm[piv * N + c] = tm; }
      float tm = y[col]; y[col] = y[piv]; y[piv] = tm;
    }
    float inv = 1.0f / Hm[col * N + col];
    for (int r = 0; r < N; ++r) {
      if (r == col) continue;
      float f = Hm[r * N + col] * inv;
      for (int c = 0; c < N; ++c) Hm[r * N + c] -= f * Hm[col * N + c];
      y[r] -= f * y[col];
    }
  }
  for (int j = 0; j < n; ++j) alpha[b * 5 + j] = y[j + 1] / Hm[(j + 1) * N + (j + 1)];
}

// xnew = sum_j alpha_j * F_j  (beta == 1.0)
__global__ void k_comb(const float* __restrict__ Fm, const float* __restrict__ alpha, int n,
                       float* __restrict__ Xs) {
  long t = (long)blockIdx.x * blockDim.x + threadIdx.x;
  if (t >= dT) return;
  long b = t / dB;
  float a = 0.f;
  for (int j = 0; j < n; ++j) a += alpha[b * 5 + j] * Fm[(long)j * dT + t];
  Xs[t] = a;
}

// ---------------- head helpers ----------------
__global__ void k_bn_stats(const float* __restrict__ z, float* mu, float* rs) {
  int c = blockIdx.x;
  float s = 0.f, s2 = 0.f;
  for (long e = threadIdx.x; e < (long)Bn * Pp; e += 256) {
    long b = e / Pp, p = e % Pp;
    float v = z[(b * Cc + c) * Pp + p];
    s += v; s2 += v * v;
  }
  float S = blk_sum256(s), S2 = blk_sum256(s2);
  if (threadIdx.x == 0) {
    float m = S / (float)((long)Bn * Pp);
    float var = S2 / (float)((long)Bn * Pp) - m * m;
    mu[c] = m; rs[c] = rsqrtf(var + EPSV);
  }
}

__global__ void k_fold(const float* __restrict__ f1w, const float* __restrict__ f1b,
                       const float* __restrict__ bw, const float* __restrict__ bb,
                       const float* __restrict__ mu, const float* __restrict__ rs,
                       float* __restrict__ fw, float* __restrict__ fb) {
  int o = threadIdx.x;
  if (o >= 32) return;
  float acc = f1b[o];
  for (int c = 0; c < 32; ++c) {
    float a = bw[c] * rs[c];
    float bc = bb[c] - a * mu[c];
    fw[o * 32 + c] = f1w[o * 32 + c] * a;
    acc += f1w[o * 32 + c] * bc;
  }
  fb[o] = acc;
}

// ---------------- host-side f_block ----------------
struct FB {
  const float *sw1r, *sw1i, *sw2r, *sw2i, *w0w, *w0b, *n1w, *n1b, *n2w, *n2b;
  const float* xi;
  float *Zy, *Zf, *Om, *Tt, *cwt, *swt, *cht, *sht, *yb, *t1, *gm, *gr;
  hipStream_t st;
};

static void run_fblock(const FB& f, const float* zin, float* zout) {
  k_dftw<<<1152, 256, 0, f.st>>>(zin, f.cwt, f.swt, f.Zy);
  k_dfth<<<144, 256, 0, f.st>>>(f.Zy, f.cht, f.sht, f.Zf);
  k_mix<<<144, 256, 0, f.st>>>(f.Zf, f.sw1r, f.sw1i, f.sw2r, f.sw2i, f.Om);
  k_synw<<<2304, 256, 0, f.st>>>(f.Om, f.cwt, f.swt, f.Tt);
  k_lin32_wmma<<<2304, 128, 0, f.st>>>(zin, f.w0w, f.w0b, f.yb);           // y2
  k_synh<<<18432, 256, 0, f.st>>>(f.Tt, f.cht, f.sht, f.yb, f.xi, f.t1);   // t1 = xi + gelu(y1+y2)
  k_gn_stats<<<8, 256, 0, f.st>>>(f.t1, 16, 2, f.gm, f.gr);
  k_resid<<<18432, 256, 0, f.st>>>(zin, f.t1, f.gm, f.gr, f.n1w, f.n1b, f.yb); // t2 reuses yb
  k_gn_stats<<<8, 256, 0, f.st>>>(f.yb, 16, 2, f.gm, f.gr);
  k_gnapply<<<18432, 256, 0, f.st>>>(f.yb, f.gm, f.gr, f.n2w, f.n2b, zout);
}

extern "C" void kernel_launch(void* const* d_in, const int* in_sizes, int n_in,
                              void* d_out, int out_size, void* d_ws, size_t ws_size,
                              hipStream_t stream) {
  (void)in_sizes; (void)n_in; (void)out_size; (void)ws_size;
  const float* x      = (const float*)d_in[0];
  const float* fc0_w  = (const float*)d_in[1];
  const float* fc0_b  = (const float*)d_in[2];
  const float* conv_w = (const float*)d_in[3];
  const float* injw   = (const float*)d_in[4];
  const float* injb   = (const float*)d_in[5];
  const float* q_w    = (const float*)d_in[6];
  const float* q_b    = (const float*)d_in[7];
  const float* sw1r   = (const float*)d_in[8];
  const float* sw1i   = (const float*)d_in[9];
  const float* sw2r   = (const float*)d_in[10];
  const float* sw2i   = (const float*)d_in[11];
  const float* w0w    = (const float*)d_in[12];
  const float* w0b    = (const float*)d_in[13];
  const float* n1w    = (const float*)d_in[14];
  const float* n1b    = (const float*)d_in[15];
  const float* n2w    = (const float*)d_in[16];
  const float* n2b    = (const float*)d_in[17];
  const float* bnw    = (const float*)d_in[18];
  const float* bnb    = (const float*)d_in[19];
  const float* fc1w   = (const float*)d_in[20];
  const float* fc1b   = (const float*)d_in[21];
  const float* fc2w   = (const float*)d_in[22];
  const float* fc2b   = (const float*)d_in[23];

  // workspace carve (floats)
  float* w = (float*)d_ws;
  float* xin = w;  w += dT;
  float* xi  = w;  w += dT;
  float* X   = w;  w += 5 * dT;
  float* Fm  = w;  w += 5 * dT;
  float* t1  = w;  w += dT;          // also conv16 scratch (needs dT/2)
  float* yb  = w;  w += dT;          // y2, then t2
  float* Zy  = w;  w += (long)Bn * Cc * Hh * NM * 2;     // 589824
  float* Zf  = w;  w += (long)Bn * Cc * NKX * NM * 2;    // 73728
  float* Om  = w;  w += (long)Bn * Cc * NKX * NM * 2;    // 73728
  float* Tt  = w;  w += (long)Bn * Cc * NKX * Wp * 2;    // 1179648
  float* cwt = w;  w += NM * Wp;
  float* swt = w;  w += NM * Wp;
  float* cht = w;  w += NKX * Wp;
  float* sht = w;  w += NKX * Wp;
  float* gm  = w;  w += 16;
  float* gr  = w;  w += 16;
  float* bnMu = w; w += 32;
  float* bnR  = w; w += 32;
  float* fw   = w; w += 1024;
  float* fb   = w; w += 32;
  float* gramP = w; w += (long)GRAM_CH * Bn * 16;
  float* alpha = w; w += 32;

  k_tables<<<18, 256, 0, stream>>>(cwt, swt, cht, sht);

  // lift + injection (xi is constant across the DEQ iteration)
  k_fc0<<<18432, 256, 0, stream>>>(x, fc0_w, fc0_b, xin);
  k_conv3<<<9216, 256, 0, stream>>>(xin, conv_w, t1);
  k_gn_stats<<<8, 256, 0, stream>>>(t1, 8, 2, gm, gr);
  k_injq<<<18432, 256, 0, stream>>>(t1, gm, gr, injw, injb, q_w, q_b, xi);

  FB f;
  f.sw1r = sw1r; f.sw1i = sw1i; f.sw2r = sw2r; f.sw2i = sw2i;
  f.w0w = w0w; f.w0b = w0b; f.n1w = n1w; f.n1b = n1b; f.n2w = n2w; f.n2b = n2b;
  f.xi = xi; f.Zy = Zy; f.Zf = Zf; f.Om = Om; f.Tt = Tt;
  f.cwt = cwt; f.swt = swt; f.cht = cht; f.sht = sht;
  f.yb = yb; f.t1 = t1; f.gm = gm; f.gr = gr; f.st = stream;

  // Anderson: X0=0; F0=f(X0); X1=F0; F1=f(F0)
  k_fill0<<<18432, 256, 0, stream>>>(X, dT);
  run_fblock(f, X, Fm);
  k_copy<<<18432, 256, 0, stream>>>(Fm, X + dT, dT);
  run_fblock(f, Fm, Fm + dT);

  for (int k = 2; k < 16; ++k) {
    int n = k < 5 ? k : 5;
    dim3 gg(GRAM_CH, Bn);
    k_gram<<<gg, 256, 0, stream>>>(Fm, X, n, gramP);
    k_solve<<<1, 32, 0, stream>>>(gramP, n, alpha);
    k_comb<<<18432, 256, 0, stream>>>(Fm, alpha, n, X + (long)(k % 5) * dT);
    run_fblock(f, X + (long)(k % 5) * dT, Fm + (long)(k % 5) * dT);
  }
  // last = 15 % 5 = 0: z_star = X[0], final z = f_block(z_star) == Fm slot 0
  const float* z = Fm;

  // head: BN (batch stats) folded into fc1, gelu, fc2 — fused WMMA kernel
  k_bn_stats<<<32, 256, 0, stream>>>(z, bnMu, bnR);
  k_fold<<<1, 32, 0, stream>>>(fc1w, fc1b, bnw, bnb, bnMu, bnR, fw, fb);
  k_head_wmma<<<2304, 128, 0, stream>>>(z, fw, fb, fc2w, fc2b, (float*)d_out);
}